// SimpleRNN_34548716929051
// MI455X (gfx1250) — compile-verified
//
#include <hip/hip_runtime.h>
#include <math.h>

// SimpleRNN, restructured as parallel GEMMs over the linear recurrence.
// h_final = sum_t c_t * A^(2047-t),  A = Wh^T,  c_t = wx[t] (+bh for t>0).

typedef float v2f __attribute__((ext_vector_type(2)));
typedef float v8f __attribute__((ext_vector_type(8)));

#define SEQN 2048
#define BSZ  256
#define XD   128
#define HD   256
#define SB   (SEQN * BSZ)   // 524288 rows for the wx GEMM
#define HH   (HD * HD)      // 65536 elements per 256x256 matrix
#define CHUNK 128
#define NCHUNK (SEQN / CHUNK)  // 16

// CDNA5 async global->LDS staging (ASYNCcnt). The clang builtins use CUDA
// LangAS-qualified pointer params that HIP source cannot name, so we emit the
// instructions with inline asm (per the bridge doc's guidance). Flip to 0 to
// use the synchronous LDS fallback.
#ifndef USE_ASYNC_LDS_ASM
#define USE_ASYNC_LDS_ASM 1
#endif

// ---- f32 WMMA micro-kernel helpers (wave32, one 16x16 tile per wave) ----

__device__ __forceinline__ v8f wmma4(v2f a, v2f b, v8f c) {
  // D = A(16x4) * B(4x16) + C(16x16), all f32
  return __builtin_amdgcn_wmma_f32_16x16x4_f32(
      /*neg_a=*/false, a, /*neg_b=*/false, b,
      /*c_mod=*/(short)0, c, /*reuse_a=*/false, /*reuse_b=*/false);
}

// A fragment: A row-major, base pre-offset to tile row 0, cols k0..k0+3.
__device__ __forceinline__ v2f frag_a(const float* A, size_t lda, int k0) {
  int lane = threadIdx.x & 31;
  int m = lane & 15;
  int k = k0 + ((lane >> 4) << 1);
  const float* p = A + (size_t)m * lda + k;
  v2f r; r.x = p[0]; r.y = p[1];
  return r;
}

// B fragment: B row-major B[k][n], base pre-offset to col n0.
__device__ __forceinline__ v2f frag_b(const float* B, size_t ldb, int k0) {
  int lane = threadIdx.x & 31;
  int n = lane & 15;
  int k = k0 + ((lane >> 4) << 1);
  const float* p = B + (size_t)k * ldb + n;
  v2f r; r.x = p[0]; r.y = p[ldb];
  return r;
}

// B fragment from transposed storage Bt[n][k] (i.e. B[k][n] = Bt[n*ldt+k]).
__device__ __forceinline__ v2f frag_bt(const float* Bt, size_t ldt, int k0) {
  int lane = threadIdx.x & 31;
  int n = lane & 15;
  int k = k0 + ((lane >> 4) << 1);
  const float* p = Bt + (size_t)n * ldt + k;
  v2f r; r.x = p[0]; r.y = p[1];
  return r;
}

__device__ __forceinline__ v8f frag_c(const float* C, size_t ldc) {
  int lane = threadIdx.x & 31;
  int n = lane & 15;
  int g = lane >> 4;
  v8f r;
#pragma unroll
  for (int v = 0; v < 8; ++v) r[v] = C[(size_t)(v + 8 * g) * ldc + n];
  return r;
}

__device__ __forceinline__ void store_c(float* C, size_t ldc, v8f acc) {
  int lane = threadIdx.x & 31;
  int n = lane & 15;
  int g = lane >> 4;
#pragma unroll
  for (int v = 0; v < 8; ++v) C[(size_t)(v + 8 * g) * ldc + n] = acc[v];
}

// LDS fragment: slab holds K-pairs: slab[kp*LD2 + col] = {M[2kp][col], M[2kp+1][col]}
// => one ds_load_b64 per fragment. LD2=80 puts the two lane-groups (rows kp,
// kp+1) on disjoint bank halves (80*2 dwords = 160 ≡ 32 mod 64).
#define LD2 80
__device__ __forceinline__ v2f frag_lds(const v2f* slab, int k0) {
  int lane = threadIdx.x & 31;
  int c = lane & 15;
  int kp = (k0 >> 1) + (lane >> 4);
  return slab[kp * LD2 + c];
}

// ---- Kernel 1: WX'[t,b,:] = x[t,b,:] @ Wx^T + bx (+ bh for t>0) ----
__global__ void wx_kernel(const float* __restrict__ x, const float* __restrict__ Wx,
                          const float* __restrict__ bx, const float* __restrict__ bh,
                          float* __restrict__ WXp) {
  int wave = threadIdx.x >> 5;
  size_t rtile = (size_t)blockIdx.x * 4 + wave;
  size_t r0 = rtile * 16;
  int h0 = blockIdx.y * 16;
  const float* A  = x  + r0 * XD;
  const float* Bt = Wx + (size_t)h0 * XD;  // Wx is (H x X): B[k][h] = Wx[h][k]
  v8f acc = {};
#pragma unroll
  for (int k = 0; k < XD; k += 4)
    acc = wmma4(frag_a(A, XD, k), frag_bt(Bt, XD, k), acc);

  int lane = threadIdx.x & 31;
  int n = lane & 15, g = lane >> 4;
  int t = (int)(r0 >> 8);
  float bias = bx[h0 + n] + (t > 0 ? bh[h0 + n] : 0.f);
  float* C = WXp + r0 * HD + h0;
#pragma unroll
  for (int v = 0; v < 8; ++v) C[(size_t)(v + 8 * g) * HD + n] = acc[v] + bias;
}

// ---- Kernel 2: Pow[0] = I, Pow[1] = Wh^T ----
__global__ void init_pow_kernel(const float* __restrict__ Wh, float* __restrict__ Pow) {
  int idx = blockIdx.x * blockDim.x + threadIdx.x;
  int r = idx >> 8, c = idx & 255;
  Pow[idx] = (r == c) ? 1.f : 0.f;
  Pow[HH + idx] = Wh[c * HD + r];
}

// ---- Kernel 3: log-doubling power level: Pow[p+m] = Pow[p] @ Pow[m], m=1..p ----
__global__ void pow_level_kernel(float* __restrict__ Pow, int p) {
  int m = blockIdx.z + 1;
  int h0 = blockIdx.x * 16;
  int b0 = blockIdx.y * 16;
  const float* A = Pow + (size_t)p * HH + (size_t)b0 * HD;
  const float* B = Pow + (size_t)m * HH + h0;
  v8f acc = {};
  for (int k = 0; k < HD; k += 4)
    acc = wmma4(frag_a(A, HD, k), frag_b(B, HD, k), acc);
  store_c(Pow + (size_t)(p + m) * HH + (size_t)b0 * HD + h0, HD, acc);
}

// ---- Kernel 4: chunk sums S_j = sum_i c_{128j+i} * A^(127-i) ----
// Cooperative: 8 waves, block tile 64(b) x 64(h); wave (bw,hw) -> 16x32.
// K-slabs of 32 staged to LDS (K-paired, A transposed), double buffered,
// fetched with async global->LDS loads overlapped with WMMA compute.

#if USE_ASYNC_LDS_ASM
// VDST = LDS byte offset (low 32 bits of a generic pointer to __shared__),
// VADDR = 64-bit global address in a VGPR pair.
__device__ __forceinline__ void async_b32(const void* g, void* l) {
  unsigned lds = (unsigned)(size_t)l;
  unsigned long long ga = (unsigned long long)(size_t)g;
  asm volatile("global_load_async_to_lds_b32 %0, %1, off"
               :
               : "v"(lds), "v"(ga)
               : "memory");
}
__device__ __forceinline__ void async_b64(const void* g, void* l) {
  unsigned lds = (unsigned)(size_t)l;
  unsigned long long ga = (unsigned long long)(size_t)g;
  asm volatile("global_load_async_to_lds_b64 %0, %1, off"
               :
               : "v"(lds), "v"(ga)
               : "memory");
}
__device__ __forceinline__ void wait_async0() {
  asm volatile("s_wait_asynccnt 0x0" ::: "memory");
}
#endif

// Stage one phase: A slab (64 m x 32 k, transposed+paired) and P slab
// (32 k x 64 n, paired). 256 threads cooperate; global reads coalesced.
__device__ __forceinline__ void stage_slabs(const float* __restrict__ Ag,
                                            const float* __restrict__ Pg,
                                            v2f* __restrict__ Abuf,
                                            v2f* __restrict__ Pbuf,
                                            int kbase, int tid) {
#if USE_ASYNC_LDS_ASM
  {  // A: global pair (contiguous 8B) -> LDS element, b64 async
    int kp = tid & 15;
    int mg = tid >> 4;  // 0..15
#pragma unroll
    for (int r = 0; r < 4; ++r) {
      int m = mg * 4 + r;
      async_b64(Ag + (size_t)m * HD + kbase + 2 * kp, &Abuf[kp * LD2 + m]);
    }
  }
  {  // P: pair spans two global rows -> two b32 asyncs
    int c = tid & 63;
    int kg = tid >> 6;  // 0..3
#pragma unroll
    for (int q = 0; q < 4; ++q) {
      int kp = kg * 4 + q;
      const float* g0 = Pg + (size_t)(kbase + 2 * kp) * HD + c;
      float* l = (float*)&Pbuf[kp * LD2 + c];
      async_b32(g0, l);
      async_b32(g0 + HD, l + 1);
    }
  }
#else
  {
    int kp = tid & 15;
    int mg = tid >> 4;
#pragma unroll
    for (int r = 0; r < 4; ++r) {
      int m = mg * 4 + r;
      const float* p = Ag + (size_t)m * HD + kbase + 2 * kp;
      v2f v; v.x = p[0]; v.y = p[1];
      Abuf[kp * LD2 + m] = v;
    }
  }
  {
    int c = tid & 63;
    int kg = tid >> 6;
#pragma unroll
    for (int q = 0; q < 4; ++q) {
      int kp = kg * 4 + q;
      const float* g0 = Pg + (size_t)(kbase + 2 * kp) * HD + c;
      v2f v; v.x = g0[0]; v.y = g0[HD];
      Pbuf[kp * LD2 + c] = v;
    }
  }
#endif
}

__global__ void __launch_bounds__(256)
chunk_kernel(const float* __restrict__ WXp, const float* __restrict__ Pow,
             float* __restrict__ S) {
  __shared__ v2f At[2][16 * LD2];  // 2 x 10.0 KB
  __shared__ v2f Pt[2][16 * LD2];  // 2 x 10.0 KB
  int j = blockIdx.z;
  int b0 = blockIdx.y * 64;
  int h0 = blockIdx.x * 64;
  int tid = threadIdx.x;
  int w = tid >> 5, bw = w & 3, hw = w >> 2;

  v8f acc0 = {}, acc1 = {};
  const int NPH = CHUNK * (HD / 32);  // 1024 phases (i x k-slab)

#if USE_ASYNC_LDS_ASM
  {  // prologue: stage phase 0 into buffer 0
    const float* Ag = WXp + ((size_t)(j * CHUNK + 0) * BSZ + b0) * HD;
    const float* Pg = Pow + (size_t)(CHUNK - 1 - 0) * HH + h0;
    stage_slabs(Ag, Pg, At[0], Pt[0], 0, tid);
    wait_async0();
  }
  __syncthreads();
  for (int ph = 0; ph < NPH; ++ph) {
    int buf = ph & 1;
    if (ph + 1 < NPH) {  // fire async fetch of next phase into other buffer
      int i = (ph + 1) >> 3, ks = (ph + 1) & 7;
      const float* Ag = WXp + ((size_t)(j * CHUNK + i) * BSZ + b0) * HD;
      const float* Pg = Pow + (size_t)(CHUNK - 1 - i) * HH + h0;
      stage_slabs(Ag, Pg, At[buf ^ 1], Pt[buf ^ 1], ks * 32, tid);
    }
    const v2f* As = &At[buf][0] + 16 * bw;
    const v2f* Ps = &Pt[buf][0] + 32 * hw;
#pragma unroll
    for (int k0 = 0; k0 < 32; k0 += 4) {
      v2f a = frag_lds(As, k0);
      acc0 = wmma4(a, frag_lds(Ps, k0), acc0);
      acc1 = wmma4(a, frag_lds(Ps + 16, k0), acc1);
    }
    wait_async0();   // next buffer's async writes have landed in LDS
    __syncthreads();
  }
#else
  for (int ph = 0; ph < NPH; ++ph) {
    int i = ph >> 3, ks = ph & 7;
    const float* Ag = WXp + ((size_t)(j * CHUNK + i) * BSZ + b0) * HD;
    const float* Pg = Pow + (size_t)(CHUNK - 1 - i) * HH + h0;
    stage_slabs(Ag, Pg, At[0], Pt[0], ks * 32, tid);
    __syncthreads();
    const v2f* As = &At[0][0] + 16 * bw;
    const v2f* Ps = &Pt[0][0] + 32 * hw;
#pragma unroll
    for (int k0 = 0; k0 < 32; k0 += 4) {
      v2f a = frag_lds(As, k0);
      acc0 = wmma4(a, frag_lds(Ps, k0), acc0);
      acc1 = wmma4(a, frag_lds(Ps + 16, k0), acc1);
    }
    __syncthreads();
  }
#endif

  float* C = S + (size_t)j * HH + (size_t)(b0 + 16 * bw) * HD + h0 + 32 * hw;
  store_c(C, HD, acc0);
  store_c(C + 16, HD, acc1);
}

// ---- Kernel 5: C = A @ B + D (256x256), optional tanh on output ----
__global__ void matmul_add_kernel(const float* __restrict__ A, const float* __restrict__ B,
                                  const float* __restrict__ D, float* __restrict__ C,
                                  int apply_tanh) {
  int h0 = blockIdx.x * 16;
  int b0 = blockIdx.y * 16;
  v8f acc = frag_c(D + (size_t)b0 * HD + h0, HD);
  const float* Ab = A + (size_t)b0 * HD;
  const float* Bb = B + h0;
  for (int k = 0; k < HD; k += 4)
    acc = wmma4(frag_a(Ab, HD, k), frag_b(Bb, HD, k), acc);
  if (apply_tanh) {
#pragma unroll
    for (int v = 0; v < 8; ++v) acc[v] = tanhf(acc[v]);
  }
  store_c(C + (size_t)b0 * HD + h0, HD, acc);
}

// ---- Kernel 6: tiny MLP head ----
__global__ void head_kernel(const float* __restrict__ Ht,
                            const float* __restrict__ W1, const float* __restrict__ b1,
                            const float* __restrict__ W2, const float* __restrict__ b2,
                            const float* __restrict__ Wo, const float* __restrict__ bo,
                            float* __restrict__ out) {
  int b = threadIdx.x;
  float f1[40];
  for (int o = 0; o < 40; ++o) {
    float s = b1[o];
    for (int k = 0; k < HD; ++k) s = fmaf(Ht[b * HD + k], W1[o * HD + k], s);
    f1[o] = fmaxf(s, 0.f);
  }
  float f2[10];
  for (int o = 0; o < 10; ++o) {
    float s = b2[o];
    for (int k = 0; k < 40; ++k) s = fmaf(f1[k], W2[o * 40 + k], s);
    f2[o] = fmaxf(s, 0.f);
  }
  for (int o = 0; o < 2; ++o) {
    float s = bo[o];
    for (int k = 0; k < 10; ++k) s = fmaf(f2[k], Wo[o * 10 + k], s);
    out[b * 2 + o] = s;
  }
}

extern "C" void kernel_launch(void* const* d_in, const int* in_sizes, int n_in,
                              void* d_out, int out_size, void* d_ws, size_t ws_size,
                              hipStream_t stream) {
  (void)in_sizes; (void)n_in; (void)out_size; (void)ws_size;
  const float* x  = (const float*)d_in[0];
  const float* Wx = (const float*)d_in[1];
  const float* bx = (const float*)d_in[2];
  const float* Wh = (const float*)d_in[3];
  const float* bh = (const float*)d_in[4];
  const float* W1 = (const float*)d_in[5];
  const float* b1 = (const float*)d_in[6];
  const float* W2 = (const float*)d_in[7];
  const float* b2 = (const float*)d_in[8];
  const float* Wo = (const float*)d_in[9];
  const float* bo = (const float*)d_in[10];
  float* out = (float*)d_out;

  // Workspace layout (floats): WX' | Pow[0..128] | S[0..15] | Hb0 | Hb1
  float* ws  = (float*)d_ws;
  float* WXp = ws;
  float* Pow = WXp + (size_t)SB * HD;
  float* S   = Pow + (size_t)129 * HH;
  float* Hb0 = S   + (size_t)NCHUNK * HH;
  float* Hb1 = Hb0 + HH;

  wx_kernel<<<dim3(SB / 16 / 4, HD / 16), 128, 0, stream>>>(x, Wx, bx, bh, WXp);
  init_pow_kernel<<<HH / 256, 256, 0, stream>>>(Wh, Pow);
  for (int p = 1; p <= 64; p <<= 1)
    pow_level_kernel<<<dim3(HD / 16, HD / 16, p), 32, 0, stream>>>(Pow, p);

  chunk_kernel<<<dim3(HD / 64, BSZ / 64, NCHUNK), 256, 0, stream>>>(WXp, Pow, S);

  const float* P128 = Pow + (size_t)128 * HH;
  const float* cur = S;
  for (int j = 1; j < NCHUNK; ++j) {
    float* nxt = (j & 1) ? Hb0 : Hb1;
    matmul_add_kernel<<<dim3(HD / 16, BSZ / 16), 32, 0, stream>>>(
        cur, P128, S + (size_t)j * HH, nxt, (j == NCHUNK - 1) ? 1 : 0);
    cur = nxt;
  }
  head_kernel<<<1, BSZ, 0, stream>>>(cur, W1, b1, W2, b2, Wo, bo, out);
}